// DGCNN_Transformer_73778948211308
// MI455X (gfx1250) — compile-verified
//
#include <hip/hip_runtime.h>

typedef float v2f __attribute__((ext_vector_type(2)));
typedef float v8f __attribute__((ext_vector_type(8)));

#define EPSV 1e-5f

// =====================================================================
// Generic strided/batched WMMA GEMM (fp32, V_WMMA_F32_16X16X4_F32).
// C[b](m,n) = sum_k A[b](m,k)*B[b](k,n)  (+bias[m]) (optional relu)
// One wave (32 threads) computes a 16x32 tile (two 16x16 accumulators,
// A operand shared). Full tiles take a branch-free pointer-increment
// path; edge tiles use clamped indices + selects (no EXEC divergence,
// EXEC is all-ones at every WMMA as required by the ISA).
// A lanes 0-15: M=lane, K = k0+0/1 ; lanes 16-31: M=lane-16, K = k0+2/3
// B lanes 0-15: N=lane, K = k0+0/1 ; lanes 16-31: N=lane-16, K = k0+2/3
// C lane l<16 : col N=l,  rows M = 0..7 ; l>=16: col N=l-16, rows M=8..15
// =====================================================================
__global__ void wmma_gemm_k(const float* __restrict__ A, const float* __restrict__ Bm,
                            const float* __restrict__ bias, float* __restrict__ C,
                            int M, int N, int Kd,
                            int sAm, int sAk, int sAb,
                            int sBk, int sBn, int sBb,
                            int sCm, int sCn, int sCb,
                            int act)
{
  int tm = blockIdx.x * 16;
  int tn = blockIdx.y * 32;
  int b  = blockIdx.z;
  int lane = threadIdx.x & 31;
  int half = lane >> 4;
  int l    = lane & 15;
  const float* Ab = A  + b * sAb;
  const float* Bb = Bm + b * sBb;
  float*       Cb = C  + b * sCb;
  int am  = tm + l;
  int bn0 = tn + l;
  int bn1 = bn0 + 16;
  v8f acc0 = {}, acc1 = {};
  bool full = (tm + 16 <= M) && (tn + 32 <= N) && ((Kd & 3) == 0);
  if (full) {
    const float* ap  = Ab + am * sAm + (half * 2) * sAk;
    const float* bp0 = Bb + (half * 2) * sBk + bn0 * sBn;
    const float* bp1 = Bb + (half * 2) * sBk + bn1 * sBn;
    int aStep = 4 * sAk, bStep = 4 * sBk;
    for (int k0 = 0; k0 < Kd; k0 += 4) {
      v2f av, bv0, bv1;
      av.x  = ap[0];  av.y  = ap[sAk];
      bv0.x = bp0[0]; bv0.y = bp0[sBk];
      bv1.x = bp1[0]; bv1.y = bp1[sBk];
      acc0 = __builtin_amdgcn_wmma_f32_16x16x4_f32(false, av, false, bv0, (short)0, acc0, false, false);
      acc1 = __builtin_amdgcn_wmma_f32_16x16x4_f32(false, av, false, bv1, (short)0, acc1, false, false);
      ap += aStep; bp0 += bStep; bp1 += bStep;
    }
  } else {
    int amc = am  < M ? am  : 0;
    int bc0 = bn0 < N ? bn0 : 0;
    int bc1 = bn1 < N ? bn1 : 0;
    bool mv  = am  < M;
    bool n0v = bn0 < N;
    bool n1v = bn1 < N;
    for (int k0 = 0; k0 < Kd; k0 += 4) {
      int ka   = k0 + half * 2;
      bool k0v = ka     < Kd;
      bool k1v = (ka+1) < Kd;
      int ka0  = k0v ? ka     : 0;
      int ka1  = k1v ? ka + 1 : 0;
      float a0  = Ab[amc * sAm + ka0 * sAk];
      float a1  = Ab[amc * sAm + ka1 * sAk];
      float b00 = Bb[ka0 * sBk + bc0 * sBn];
      float b01 = Bb[ka1 * sBk + bc0 * sBn];
      float b10 = Bb[ka0 * sBk + bc1 * sBn];
      float b11 = Bb[ka1 * sBk + bc1 * sBn];
      v2f av, bv0, bv1;
      av.x  = (mv  && k0v) ? a0  : 0.f;
      av.y  = (mv  && k1v) ? a1  : 0.f;
      bv0.x = (n0v && k0v) ? b00 : 0.f;
      bv0.y = (n0v && k1v) ? b01 : 0.f;
      bv1.x = (n1v && k0v) ? b10 : 0.f;
      bv1.y = (n1v && k1v) ? b11 : 0.f;
      acc0 = __builtin_amdgcn_wmma_f32_16x16x4_f32(false, av, false, bv0, (short)0, acc0, false, false);
      acc1 = __builtin_amdgcn_wmma_f32_16x16x4_f32(false, av, false, bv1, (short)0, acc1, false, false);
    }
  }
  for (int r = 0; r < 8; ++r) {
    int m = tm + half * 8 + r;
    if (m < M) {
      float v0 = acc0[r], v1 = acc1[r];
      if (bias) { float bm = bias[m]; v0 += bm; v1 += bm; }
      if (act)  { v0 = fmaxf(v0, 0.f); v1 = fmaxf(v1, 0.f); }
      if (bn0 < N) Cb[m * sCm + bn0 * sCn] = v0;
      if (bn1 < N) Cb[m * sCm + bn1 * sCn] = v1;
    }
  }
}

// =====================================================================
// EdgeConv WMMA GEMM with the graph-feature gather fused into the
// B-operand load (never materializes the (B,2C,N,K) edge tensor).
// Gather is branch-free: select-computed addresses + 0/1 multiplier.
// mode 0: accumulate per-channel sum / sumsq (for BN stats)
// mode 1: v = relu(h*scale[o]+shift[o]); atomic int-max into out[b,o,n]
// All tiles are full (O%16==0, (N*K)%16==0, 2*cin%4==0): no guards.
// =====================================================================
__global__ void edgeconv_gemm_k(const float* __restrict__ W, const float* __restrict__ X,
                                const int* __restrict__ idx,
                                int O, int cin, int Nn, int Kk, int mode,
                                float* __restrict__ sum, float* __restrict__ sumsq,
                                const float* __restrict__ scale, const float* __restrict__ shift,
                                float* __restrict__ out)
{
  __shared__ float ssum[16], ssq[16];
  int tm = blockIdx.x * 16, tj = blockIdx.y * 16, b = blockIdx.z;
  int lane = threadIdx.x & 31, half = lane >> 4, l = lane & 15;
  int K2 = 2 * cin;
  const float* Xb = X + b * cin * Nn;
  int am   = tm + l;
  int j    = tj + l;            // flat (n,kk) column
  int n_pt = j / Kk;
  int kk   = j - n_pt * Kk;
  int nbr  = idx[(b * Nn + n_pt) * Kk + kk];
  const float* ap = W + am * K2 + half * 2;   // even offset -> 8B aligned
  v8f acc = {};
  for (int k0 = 0; k0 < K2; k0 += 4) {
    int c = k0 + half * 2;
    v2f av = *reinterpret_cast<const v2f*>(ap);
    ap += 4;
    v2f bv;
    {
      bool lo = c < cin;
      int cc   = lo ? c : (c - cin);
      int base = cc * Nn;
      float w  = lo ? 1.f : 0.f;
      float vn = Xb[base + (lo ? nbr : n_pt)];
      float vc = Xb[base + n_pt];
      bv.x = vn - w * vc;
    }
    {
      int c1 = c + 1;
      bool lo = c1 < cin;
      int cc   = lo ? c1 : (c1 - cin);
      int base = cc * Nn;
      float w  = lo ? 1.f : 0.f;
      float vn = Xb[base + (lo ? nbr : n_pt)];
      float vc = Xb[base + n_pt];
      bv.y = vn - w * vc;
    }
    acc = __builtin_amdgcn_wmma_f32_16x16x4_f32(false, av, false, bv, (short)0, acc, false, false);
  }
  if (mode == 0) {
    if (lane < 16) { ssum[lane] = 0.f; ssq[lane] = 0.f; }
    __syncthreads();
    for (int r = 0; r < 8; ++r) {
      float v = acc[r];
      int ml = half * 8 + r;
      atomicAdd(&ssum[ml], v);
      atomicAdd(&ssq[ml], v * v);
    }
    __syncthreads();
    if (lane < 16) {
      atomicAdd(&sum[tm + lane],   ssum[lane]);
      atomicAdd(&sumsq[tm + lane], ssq[lane]);
    }
  } else {
    for (int r = 0; r < 8; ++r) {
      int m = tm + half * 8 + r;
      float v = fmaxf(acc[r] * scale[m] + shift[m], 0.f);
      atomicMax((int*)&out[(b * O + m) * Nn + n_pt], __float_as_int(v));
    }
  }
}

// ---------------- small helper kernels ----------------
__global__ void fillf_k(float* p, float v, long long n) {
  long long i = (long long)blockIdx.x * blockDim.x + threadIdx.x;
  if (i < n) p[i] = v;
}
__global__ void fillb_k(unsigned char* p, unsigned char v, long long n) {
  long long i = (long long)blockIdx.x * blockDim.x + threadIdx.x;
  if (i < n) p[i] = v;
}
// (B,N,C) -> (B,C,N)
__global__ void transpose_k(const float* __restrict__ x, float* __restrict__ xc,
                            int Bc, int Nn, int C) {
  long long i = (long long)blockIdx.x * blockDim.x + threadIdx.x;
  if (i >= (long long)Bc * Nn * C) return;
  int c = (int)(i % C);
  int n = (int)((i / C) % Nn);
  int b = (int)(i / ((long long)C * Nn));
  xc[(b * C + c) * Nn + n] = x[i];
}
// BN stats over (b, r) per channel o; layout h[b*sb + o*so + r]
__global__ void bn_stats_k(const float* __restrict__ h, const float* __restrict__ g,
                           const float* __restrict__ bb, float* scale, float* shift,
                           int O, int R, int Bc, int so, int sb) {
  __shared__ float rs[256], rq[256];
  int o = blockIdx.x, t = threadIdx.x;
  float s = 0.f, s2 = 0.f;
  for (int b = 0; b < Bc; ++b) {
    const float* hp = h + b * sb + o * so;
    for (int r = t; r < R; r += blockDim.x) { float v = hp[r]; s += v; s2 += v * v; }
  }
  rs[t] = s; rq[t] = s2; __syncthreads();
  for (int k = 128; k > 0; k >>= 1) { if (t < k) { rs[t] += rs[t + k]; rq[t] += rq[t + k]; } __syncthreads(); }
  if (t == 0) {
    float cnt  = (float)R * (float)Bc;
    float mean = rs[0] / cnt;
    float var  = rq[0] / cnt - mean * mean;
    float sc   = g[o] * rsqrtf(var + EPSV);
    scale[o] = sc; shift[o] = bb[o] - mean * sc;
  }
}
__global__ void bn_apply_relu_k(float* h, const float* scale, const float* shift,
                                int O, int R, long long total) {
  long long i = (long long)blockIdx.x * blockDim.x + threadIdx.x;
  if (i >= total) return;
  int o = (int)((i / R) % O);
  h[i] = fmaxf(h[i] * scale[o] + shift[o], 0.f);
}
__global__ void maxpool_k(const float* __restrict__ h, float* pool, int total, int Nn) {
  int i = blockIdx.x * blockDim.x + threadIdx.x;
  if (i >= total) return;
  const float* p = h + (long long)i * Nn;
  float m = p[0];
  for (int n = 1; n < Nn; ++n) m = fmaxf(m, p[n]);
  pool[i] = m;
}
__global__ void add_eye_k(const float* t9, float* trans, int Bc) {
  int i = blockIdx.x * blockDim.x + threadIdx.x;
  if (i >= Bc * 9) return;
  int j = i % 9;
  trans[i] = t9[i] + ((j == 0 || j == 4 || j == 8) ? 1.f : 0.f);
}
__global__ void apply_stn_k(const float* __restrict__ xc, const float* __restrict__ trans,
                            float* __restrict__ xc2, int Bc, int Nn) {
  long long i = (long long)blockIdx.x * blockDim.x + threadIdx.x;
  if (i >= (long long)Bc * 10 * Nn) return;
  int n = (int)(i % Nn);
  int c = (int)((i / Nn) % 10);
  int b = (int)(i / ((long long)10 * Nn));
  if (c < 3) {
    const float* tr = trans + b * 9;
    float s = 0.f;
    for (int j = 0; j < 3; ++j) s += tr[c * 3 + j] * xc[(b * 10 + j) * Nn + n];
    xc2[i] = s;
  } else {
    xc2[i] = xc[i];
  }
}
// per-row 20-NN selection from gram matrix: d = g[n,n]-2g[n,m]+g[m,m]
__global__ void knn_select_k(const float* __restrict__ gram, int* __restrict__ idx,
                             int Bc, int Nn, int Kk) {
  int i = blockIdx.x * blockDim.x + threadIdx.x;
  if (i >= Bc * Nn) return;
  int b = i / Nn, n = i % Nn;
  const float* g  = gram + (long long)b * Nn * Nn;
  const float* gr = g + (long long)n * Nn;
  float gnn = gr[n];
  float bd[20]; int bi[20];
  int cnt = 0;
  for (int m = 0; m < Nn; ++m) {
    float d = gnn - 2.f * gr[m] + g[(long long)m * Nn + m];
    if (cnt < Kk) {
      int p = cnt++;
      while (p > 0 && bd[p - 1] > d) { bd[p] = bd[p - 1]; bi[p] = bi[p - 1]; --p; }
      bd[p] = d; bi[p] = m;
    } else if (d < bd[Kk - 1]) {
      int p = Kk - 1;
      while (p > 0 && bd[p - 1] > d) { bd[p] = bd[p - 1]; bi[p] = bi[p - 1]; --p; }
      bd[p] = d; bi[p] = m;
    }
  }
  int* op = idx + (long long)i * Kk;
  for (int k = 0; k < Kk; ++k) op[k] = bi[k];
}
__global__ void bn_finalize_k(const float* sum, const float* sumsq, const float* g,
                              const float* bb, float* scale, float* shift, int O, float cnt) {
  int o = blockIdx.x * blockDim.x + threadIdx.x;
  if (o >= O) return;
  float mean = sum[o] / cnt;
  float var  = sumsq[o] / cnt - mean * mean;
  float sc   = g[o] * rsqrtf(var + EPSV);
  scale[o] = sc; shift[o] = bb[o] - mean * sc;
}
__global__ void adj_scatter_k(const int* __restrict__ idx, unsigned char* adj,
                              int Bc, int Nn, int Kk) {
  long long i = (long long)blockIdx.x * blockDim.x + threadIdx.x;
  if (i >= (long long)Bc * Nn * Kk) return;
  int b = (int)(i / ((long long)Nn * Kk));
  long long rem = i % ((long long)Nn * Kk);
  int n = (int)(rem / Kk);
  int m = idx[i];
  unsigned char* ab = adj + (long long)b * Nn * Nn;
  ab[(long long)n * Nn + m] = 1;
  ab[(long long)m * Nn + n] = 1;
}
__global__ void concat_xcat_k(const float* x1, const float* x2, const float* x3,
                              const float* x4, float* xcat, int Bc, int Nn) {
  long long i = (long long)blockIdx.x * blockDim.x + threadIdx.x;
  if (i >= (long long)Bc * Nn * 512) return;
  int f = (int)(i % 512);
  int n = (int)((i / 512) % Nn);
  int b = (int)(i / ((long long)512 * Nn));
  float v;
  if      (f < 64)  v = x1[(b * 64  + f)         * Nn + n];
  else if (f < 128) v = x2[(b * 64  + (f - 64))  * Nn + n];
  else if (f < 256) v = x3[(b * 128 + (f - 128)) * Nn + n];
  else              v = x4[(b * 256 + (f - 256)) * Nn + n];
  xcat[i] = v;
}
// per-head scores: s1 = Wh . a[:fh], s2 = Wh . a[fh:]
__global__ void gat_scores_k(const float* __restrict__ Wh, const float* __restrict__ a,
                             float* s1, float* s2,
                             int Bc, int Nn, int fh, int F, int hoff, int nh, int h) {
  int i = blockIdx.x * blockDim.x + threadIdx.x;
  if (i >= Bc * Nn) return;
  int b = i / Nn, n = i % Nn;
  const float* w = Wh + (long long)(b * Nn + n) * F + hoff;
  float a1 = 0.f, a2 = 0.f;
  for (int f = 0; f < fh; ++f) { a1 += w[f] * a[f]; a2 += w[f] * a[fh + f]; }
  int o = (b * nh + h) * Nn + n;
  s1[o] = a1; s2[o] = a2;
}
// per (b,n,head) masked softmax attention row + elu
__global__ void gat_attn_k(const float* __restrict__ Wh, const float* __restrict__ s1,
                           const float* __restrict__ s2, const unsigned char* __restrict__ adj,
                           float* __restrict__ hout, int Nn, int fh, int F, int hoff,
                           int nh, int h) {
  __shared__ float red[64];
  int bn = blockIdx.x;
  int b  = bn / Nn, n = bn % Nn;
  const unsigned char* am = adj + (long long)(b * Nn + n) * Nn;
  float s1v = s1[(b * nh + h) * Nn + n];
  const float* s2r = s2 + (b * nh + h) * Nn;
  int t = threadIdx.x;
  float mx = -3.0e38f;
  for (int m = t; m < Nn; m += 64) {
    float e = s1v + s2r[m];
    e = e > 0.f ? e : 0.2f * e;
    e = am[m] ? e : -9.0e15f;
    mx = fmaxf(mx, e);
  }
  red[t] = mx; __syncthreads();
  for (int s = 32; s > 0; s >>= 1) { if (t < s) red[t] = fmaxf(red[t], red[t + s]); __syncthreads(); }
  mx = red[0]; __syncthreads();
  float se = 0.f;
  for (int m = t; m < Nn; m += 64) {
    float e = s1v + s2r[m];
    e = e > 0.f ? e : 0.2f * e;
    e = am[m] ? e : -9.0e15f;
    se += expf(e - mx);
  }
  red[t] = se; __syncthreads();
  for (int s = 32; s > 0; s >>= 1) { if (t < s) red[t] += red[t + s]; __syncthreads(); }
  se = red[0]; __syncthreads();
  if (t < fh) {
    float acc = 0.f;
    for (int m = 0; m < Nn; ++m) {
      float e = s1v + s2r[m];
      e = e > 0.f ? e : 0.2f * e;
      e = am[m] ? e : -9.0e15f;
      acc += expf(e - mx) * Wh[(long long)(b * Nn + m) * F + hoff + t];
    }
    float o = acc / se;
    o = o > 0.f ? o : (expf(o) - 1.f);
    hout[(long long)(b * Nn + n) * F + hoff + t] = o;
  }
}
__global__ void build_t_k(const float* cls, const float* g2, float* t, int Bc, int S, int D) {
  long long i = (long long)blockIdx.x * blockDim.x + threadIdx.x;
  if (i >= (long long)Bc * S * D) return;
  int f = (int)(i % D);
  int s = (int)((i / D) % S);
  int b = (int)(i / ((long long)D * S));
  t[i] = (s == 0) ? cls[f] : g2[(b * (S - 1) + (s - 1)) * D + f];
}
// multi-head attention row: one block per (b, head, query)
__global__ void mha_attn_k(const float* __restrict__ qkv, float* __restrict__ obuf,
                           int S, int D, int nh) {
  __shared__ float p[1025];
  __shared__ float red[128];
  __shared__ float qs[16];
  int blk = blockIdx.x;
  int q  = blk % S;
  int bh = blk / S;
  int h  = bh % nh;
  int b  = bh / nh;
  int dh = D / nh; // 16
  const float* base = qkv + (long long)b * S * (3 * D);
  const float* qv   = base + (long long)q * 3 * D + h * dh;
  int t = threadIdx.x;
  if (t < dh) qs[t] = qv[t];
  __syncthreads();
  float scl = rsqrtf((float)dh);
  float mx = -3.0e38f;
  for (int m = t; m < S; m += blockDim.x) {
    const float* kv = base + (long long)m * 3 * D + D + h * dh;
    float e = 0.f;
    for (int d = 0; d < dh; ++d) e += qs[d] * kv[d];
    e *= scl;
    p[m] = e;
    mx = fmaxf(mx, e);
  }
  red[t] = mx; __syncthreads();
  for (int s = 64; s > 0; s >>= 1) { if (t < s) red[t] = fmaxf(red[t], red[t + s]); __syncthreads(); }
  mx = red[0]; __syncthreads();
  float se = 0.f;
  for (int m = t; m < S; m += blockDim.x) { float w = expf(p[m] - mx); p[m] = w; se += w; }
  red[t] = se; __syncthreads();
  for (int s = 64; s > 0; s >>= 1) { if (t < s) red[t] += red[t + s]; __syncthreads(); }
  se = red[0]; __syncthreads();
  int d     = t & (dh - 1);
  int chunk = t >> 4; // 0..7 with blockDim=128
  float acc = 0.f;
  for (int m = chunk; m < S; m += 8)
    acc += p[m] * base[(long long)m * 3 * D + 2 * D + h * dh + d];
  red[t] = acc; __syncthreads();
  if (t < dh) {
    float s2 = 0.f;
    for (int c = 0; c < 8; ++c) s2 += red[c * 16 + t];
    obuf[((long long)b * S + q) * D + h * dh + t] = s2 / se;
  }
}
// x = LN(x + y), row of D=blockDim
__global__ void add_ln_k(float* __restrict__ x, const float* __restrict__ y,
                         const float* __restrict__ g, const float* __restrict__ bb, int D) {
  __shared__ float red[128];
  long long row = blockIdx.x;
  float* xr = x + row * D;
  const float* yr = y + row * D;
  int t = threadIdx.x;
  float v = xr[t] + yr[t];
  red[t] = v; __syncthreads();
  for (int s = 64; s > 0; s >>= 1) { if (t < s) red[t] += red[t + s]; __syncthreads(); }
  float mean = red[0] / (float)D; __syncthreads();
  float dv = v - mean;
  red[t] = dv * dv; __syncthreads();
  for (int s = 64; s > 0; s >>= 1) { if (t < s) red[t] += red[t + s]; __syncthreads(); }
  float var = red[0] / (float)D;
  xr[t] = dv * rsqrtf(var + EPSV) * g[t] + bb[t];
}
__global__ void final_feat_k(const float* __restrict__ t, float* feat, int Bc, int S, int D) {
  int i = blockIdx.x * blockDim.x + threadIdx.x;
  if (i >= Bc * D) return;
  int f = i % D, b = i / D;
  const float* tb = t + (long long)b * S * D;
  float s = 0.f;
  for (int r = 1; r < S; ++r) s += tb[(long long)r * D + f];
  feat[i] = tb[f] + s / (float)(S - 1);
}

// =====================================================================
// Host orchestration
// =====================================================================
static inline void launch_gemm(hipStream_t st,
    const float* A, const float* Bm, const float* bias, float* C,
    int M, int N, int Kd,
    int sAm, int sAk, int sAb,
    int sBk, int sBn, int sBb,
    int sCm, int sCn, int sCb,
    int batch, int act) {
  dim3 grid((M + 15) / 16, (N + 31) / 32, batch);
  wmma_gemm_k<<<grid, dim3(32), 0, st>>>(A, Bm, bias, C, M, N, Kd,
                                         sAm, sAk, sAb, sBk, sBn, sBb,
                                         sCm, sCn, sCb, act);
}

extern "C" void kernel_launch(void* const* d_in, const int* in_sizes, int n_in,
                              void* d_out, int out_size, void* d_ws, size_t ws_size,
                              hipStream_t stream) {
  const int Bc = 8, Nn = 1024, Kk = 20, S = 1025;
  (void)out_size; (void)ws_size; (void)n_in;

  // Input-order hedge: params pytree leaves are in sorted-key order; the top-
  // level dict is {'x', 'params'}. Detect which comes first by in_sizes[0].
  int p0, xi;
  if (in_sizes[0] == 128) { p0 = 0; xi = n_in - 1; } // params first, x last
  else                    { p0 = 1; xi = 0;        } // x first
  auto P = [&](int k) -> const float* { return (const float*)d_in[p0 + k]; };
  const float* x = (const float*)d_in[xi];
  // param leaf indices (sorted-key order within params)
  // 0:cls  1-3:ec1(w,g,b) 4-6:ec2 7-9:ec3 10-12:ec4
  // 13:fc.b1 14:fc.b2 15:fc.w1 16:fc.w2
  // 17-20:gat1.Ws 21-24:gat1.as 25:gat1.ob 26:gat1.ow
  // 27-30:gat2.Ws 31-34:gat2.as 35:gat2.ob 36:gat2.ow
  // 37:stn.b1 38:b2 39:b3 40:fc1b 41:fc1w 42:fc2b 43:fc2w 44:fc3b 45:fc3w
  // 46:g1 47:g2 48:g3 49:w1 50:w2 51:w3
  // 52..63: tr1 {inb,inw,l1b,l1w,l2b,l2w,ln1b,ln1g,ln2b,ln2g,ob,ow}, 64..75: tr2

  // ---- workspace bump allocator ----
  size_t off = 0;
  auto alloc = [&](size_t bytes) -> void* {
    void* p = (char*)d_ws + off;
    off += (bytes + 255) & ~(size_t)255;
    return p;
  };
  auto af = [&](size_t n) -> float* { return (float*)alloc(n * sizeof(float)); };

  float* xc    = af((size_t)Bc * 10 * Nn);
  float* xc2   = af((size_t)Bc * 10 * Nn);
  float* h1    = af((size_t)Bc * 64 * Nn);
  float* h2    = af((size_t)Bc * 128 * Nn);
  float* bigA  = af((size_t)Bc * Nn * Nn);   // STN h3 / gram matrices (reused)
  float* pool  = af((size_t)Bc * 1024);
  float* fch1  = af((size_t)Bc * 512);
  float* fch2  = af((size_t)Bc * 256);
  float* tr9   = af((size_t)Bc * 9);
  float* trans = af((size_t)Bc * 9);
  float* scale = af(1024);
  float* shift = af(1024);
  float* esum  = af(256);
  float* esq   = af(256);
  int*   idxb  = (int*)alloc((size_t)Bc * Nn * Kk * sizeof(int));
  float* x1    = af((size_t)Bc * 64 * Nn);
  float* x2    = af((size_t)Bc * 64 * Nn);
  float* x3    = af((size_t)Bc * 128 * Nn);
  float* x4    = af((size_t)Bc * 256 * Nn);
  float* xcat  = af((size_t)Bc * Nn * 512);
  unsigned char* adj = (unsigned char*)alloc((size_t)Bc * Nn * Nn);
  float* Wh    = af((size_t)Bc * Nn * 256);
  float* hout  = af((size_t)Bc * Nn * 256);
  float* gout1 = af((size_t)Bc * Nn * 256);
  float* gout2 = af((size_t)Bc * Nn * 128);
  float* s1b   = af((size_t)Bc * 4 * Nn);
  float* s2b   = af((size_t)Bc * 4 * Nn);
  float* tbuf  = af((size_t)Bc * S * 128);
  float* ybuf  = af((size_t)Bc * S * 128);
  float* qkvb  = af((size_t)Bc * S * 384);
  float* obuf  = af((size_t)Bc * S * 128);
  float* ffbuf = af((size_t)S * 2048);
  float* feat  = af((size_t)Bc * 128);
  float* fbuf  = af((size_t)Bc * 256);

  auto grid1 = [](long long n) { return dim3((unsigned)((n + 255) / 256)); };

  // ================= STN =================
  long long tot = (long long)Bc * Nn * 10;
  transpose_k<<<grid1(tot), 256, 0, stream>>>(x, xc, Bc, Nn, 10);

  // h1 = w1 @ xc   (B,64,N)
  launch_gemm(stream, P(49), xc, nullptr, h1, 64, Nn, 10,
              10, 1, 0,  Nn, 1, 10 * Nn,  Nn, 1, 64 * Nn, Bc, 0);
  bn_stats_k<<<64, 256, 0, stream>>>(h1, P(46), P(37), scale, shift, 64, Nn, Bc, Nn, 64 * Nn);
  tot = (long long)Bc * 64 * Nn;
  bn_apply_relu_k<<<grid1(tot), 256, 0, stream>>>(h1, scale, shift, 64, Nn, tot);

  launch_gemm(stream, P(50), h1, nullptr, h2, 128, Nn, 64,
              64, 1, 0,  Nn, 1, 64 * Nn,  Nn, 1, 128 * Nn, Bc, 0);
  bn_stats_k<<<128, 256, 0, stream>>>(h2, P(47), P(38), scale, shift, 128, Nn, Bc, Nn, 128 * Nn);
  tot = (long long)Bc * 128 * Nn;
  bn_apply_relu_k<<<grid1(tot), 256, 0, stream>>>(h2, scale, shift, 128, Nn, tot);

  launch_gemm(stream, P(51), h2, nullptr, bigA, 1024, Nn, 128,
              128, 1, 0,  Nn, 1, 128 * Nn,  Nn, 1, 1024 * Nn, Bc, 0);
  bn_stats_k<<<1024, 256, 0, stream>>>(bigA, P(48), P(39), scale, shift, 1024, Nn, Bc, Nn, 1024 * Nn);
  tot = (long long)Bc * 1024 * Nn;
  bn_apply_relu_k<<<grid1(tot), 256, 0, stream>>>(bigA, scale, shift, 1024, Nn, tot);

  maxpool_k<<<grid1(Bc * 1024), 256, 0, stream>>>(bigA, pool, Bc * 1024, Nn);
  launch_gemm(stream, P(41), pool, P(40), fch1, 512, Bc, 1024,
              1024, 1, 0,  1, 1024, 0,  1, 512, 0, 1, 1);
  launch_gemm(stream, P(43), fch1, P(42), fch2, 256, Bc, 512,
              512, 1, 0,  1, 512, 0,  1, 256, 0, 1, 1);
  launch_gemm(stream, P(45), fch2, P(44), tr9, 9, Bc, 256,
              256, 1, 0,  1, 256, 0,  1, 9, 0, 1, 0);
  add_eye_k<<<1, 128, 0, stream>>>(tr9, trans, Bc);
  tot = (long long)Bc * 10 * Nn;
  apply_stn_k<<<grid1(tot), 256, 0, stream>>>(xc, trans, xc2, Bc, Nn);

  // ================= EdgeConv x4 =================
  auto ec_stage = [&](const float* xin, int cin, int O,
                      int wI, int gI, int bI, float* outp) {
    // gram[b,n,m] = <xin[:,n], xin[:,m]>  via WMMA
    launch_gemm(stream, xin, xin, nullptr, bigA, Nn, Nn, cin,
                1, Nn, cin * Nn,  Nn, 1, cin * Nn,
                Nn, 1, Nn * Nn, Bc, 0);
    knn_select_k<<<grid1(Bc * Nn), 256, 0, stream>>>(bigA, idxb, Bc, Nn, Kk);
    fillf_k<<<1, 256, 0, stream>>>(esum, 0.f, O);
    fillf_k<<<1, 256, 0, stream>>>(esq, 0.f, O);
    dim3 eg(O / 16, (Nn * Kk) / 16, Bc);
    edgeconv_gemm_k<<<eg, 32, 0, stream>>>(P(wI), xin, idxb, O, cin, Nn, Kk, 0,
                                           esum, esq, nullptr, nullptr, nullptr);
    bn_finalize_k<<<1, 256, 0, stream>>>(esum, esq, P(gI), P(bI), scale, shift,
                                         O, (float)((long long)Bc * Nn * Kk));
    fillf_k<<<grid1((long long)Bc * O * Nn), 256, 0, stream>>>(outp, 0.f, (long long)Bc * O * Nn);
    edgeconv_gemm_k<<<eg, 32, 0, stream>>>(P(wI), xin, idxb, O, cin, Nn, Kk, 1,
                                           nullptr, nullptr, scale, shift, outp);
  };
  ec_stage(xc2, 10, 64, 1, 2, 3, x1);
  ec_stage(x1, 64, 64, 4, 5, 6, x2);
  ec_stage(x2, 64, 128, 7, 8, 9, x3);
  ec_stage(x3, 128, 256, 10, 11, 12, x4);

  // ================= adjacency from xyz kNN =================
  launch_gemm(stream, xc2, xc2, nullptr, bigA, Nn, Nn, 3,
              1, Nn, 10 * Nn,  Nn, 1, 10 * Nn,  Nn, 1, Nn * Nn, Bc, 0);
  knn_select_k<<<grid1(Bc * Nn), 256, 0, stream>>>(bigA, idxb, Bc, Nn, Kk);
  fillb_k<<<grid1((long long)Bc * Nn * Nn), 256, 0, stream>>>(adj, 0, (long long)Bc * Nn * Nn);
  adj_scatter_k<<<grid1((long long)Bc * Nn * Kk), 256, 0, stream>>>(idxb, adj, Bc, Nn, Kk);

  tot = (long long)Bc * Nn * 512;
  concat_xcat_k<<<grid1(tot), 256, 0, stream>>>(x1, x2, x3, x4, xcat, Bc, Nn);

  // ================= GAT x2 =================
  auto gat = [&](const float* xin, int fin, int fh,
                 int wI0, int aI0, int obI, int owI, float* goutp) {
    int F = 4 * fh;
    for (int h = 0; h < 4; ++h) {
      // Wh[b,n,hoff+f] = sum_c xin[b,n,c] * W[c,f]
      launch_gemm(stream, P(wI0 + h), xin, nullptr, Wh + h * fh, fh, Nn, fin,
                  1, fh, 0,  1, fin, Nn * fin,
                  1, F, Nn * F, Bc, 0);
      gat_scores_k<<<grid1(Bc * Nn), 256, 0, stream>>>(Wh, P(aI0 + h), s1b, s2b,
                                                       Bc, Nn, fh, F, h * fh, 4, h);
      gat_attn_k<<<Bc * Nn, 64, 0, stream>>>(Wh, s1b, s2b, adj, hout,
                                             Nn, fh, F, h * fh, 4, h);
    }
    launch_gemm(stream, P(owI), hout, P(obI), goutp, F, Nn, F,
                F, 1, 0,  1, F, Nn * F,  1, F, Nn * F, Bc, 0);
  };
  gat(xcat, 512, 64, 17, 21, 25, 26, gout1);
  gat(gout1, 256, 32, 27, 31, 35, 36, gout2);

  // ================= Transformer =================
  tot = (long long)Bc * S * 128;
  build_t_k<<<grid1(tot), 256, 0, stream>>>(P(0), gout2, tbuf, Bc, S, 128);

  auto enc = [&](int pb) {
    // qkv = t @ inw^T + inb
    launch_gemm(stream, P(pb + 1), tbuf, P(pb + 0), qkvb, 384, S, 128,
                128, 1, 0,  1, 128, S * 128,
                1, 384, S * 384, Bc, 0);
    mha_attn_k<<<Bc * 8 * S, 128, 0, stream>>>(qkvb, obuf, S, 128, 8);
    launch_gemm(stream, P(pb + 11), obuf, P(pb + 10), ybuf, 128, S, 128,
                128, 1, 0,  1, 128, S * 128,
                1, 128, S * 128, Bc, 0);
    add_ln_k<<<Bc * S, 128, 0, stream>>>(tbuf, ybuf, P(pb + 7), P(pb + 6), 128);
    for (int b = 0; b < Bc; ++b) {
      launch_gemm(stream, P(pb + 3), tbuf + (long long)b * S * 128, P(pb + 2), ffbuf,
                  2048, S, 128,  128, 1, 0,  1, 128, 0,  1, 2048, 0, 1, 1);
      launch_gemm(stream, P(pb + 5), ffbuf, P(pb + 4), ybuf + (long long)b * S * 128,
                  128, S, 2048,  2048, 1, 0,  1, 2048, 0,  1, 128, 0, 1, 0);
    }
    add_ln_k<<<Bc * S, 128, 0, stream>>>(tbuf, ybuf, P(pb + 9), P(pb + 8), 128);
  };
  enc(52);
  enc(64);

  final_feat_k<<<grid1(Bc * 128), 256, 0, stream>>>(tbuf, feat, Bc, S, 128);
  launch_gemm(stream, P(15), feat, P(13), fbuf, 256, Bc, 128,
              128, 1, 0,  1, 128, 0,  1, 256, 0, 1, 1);
  launch_gemm(stream, P(16), fbuf, P(14), (float*)d_out, 4, Bc, 256,
              256, 1, 0,  1, 256, 0,  1, 4, 0, 1, 0);
}